// FeaturePropagation_46145128628932
// MI455X (gfx1250) — compile-verified
//
#include <hip/hip_runtime.h>
#include <math.h>

#define B_    16
#define N_    4096
#define M_    1024
#define CIN_  256
#define CSK_  128
#define C1_   384
#define COUT_ 256
#define BN_EPS_ 1e-5f

typedef __attribute__((ext_vector_type(2))) float v2f;
typedef __attribute__((ext_vector_type(8))) float v8f;
typedef __attribute__((ext_vector_type(4))) unsigned int u32x4;
typedef __attribute__((ext_vector_type(4))) int i32x4;
typedef __attribute__((ext_vector_type(8))) int i32x8;

#if __has_builtin(__builtin_amdgcn_tensor_load_to_lds)
#define USE_TDM 1
#else
#define USE_TDM 0
#endif

// ---------------------------------------------------------------------------
// TDM: load a 64(rows) x 64(floats) tile, row stride 4096 floats, into LDS
// with 8-DWORD padding every 64 DWORDs (=> 72-float LDS pitch).
// D# per CDNA5 ISA ch.8: group0 {count, lds_addr, global_addr, type=2},
// group1 {data_size=4B, pad_enable, pad_interval=64dw(5), pad_amount=8dw(7),
//         tensor_dim0=4096, tensor_dim1=64, tile 64x64, dim0_stride=4096}.
// ---------------------------------------------------------------------------
__device__ __forceinline__ void tdm_load_tile_64x64(const float* src, void* lds_dst) {
#if USE_TDM
    const unsigned long long ga = (unsigned long long)(size_t)src;
    u32x4 g0;
    g0.x = 1u;                                    // count=1 valid descriptor
    g0.y = (unsigned)(size_t)lds_dst;             // LDS byte address (addr[31:0])
    g0.z = (unsigned)ga;                          // global_addr[31:0]
    g0.w = ((unsigned)(ga >> 32) & 0x01FFFFFFu)   // global_addr[56:32]
           | 0x80000000u;                         // type=2 ("image")
    i32x8 g1;
    g1[0] = 0x0F520000;   // data_size=2(4B) | pad_enable | pad_interval=5 | pad_amount=7
    g1[1] = 0x10000000;   // tensor_dim0=4096 (bits 79:48 -> low16 at [31:16])
    g1[2] = 0x00400000;   // tensor_dim1=64   (bits 111:80 -> low16 at [31:16])
    g1[3] = 0x00400000;   // tile_dim0=64     (bits 127:112)
    g1[4] = 64;           // tile_dim1=64     (bits 143:128); tile_dim2=0
    g1[5] = 4096;         // tensor_dim0_stride[31:0] = 4096 elements
    g1[6] = 0;
    g1[7] = 0;
    const i32x4 z4 = {0, 0, 0, 0};
#if defined(__clang_major__) && (__clang_major__ >= 23)
    const i32x8 z8 = {0, 0, 0, 0, 0, 0, 0, 0};
    __builtin_amdgcn_tensor_load_to_lds(g0, g1, z4, z4, z8, 0);
#else
    __builtin_amdgcn_tensor_load_to_lds(g0, g1, z4, z4, 0);
#endif
#else
    (void)src; (void)lds_dst;
#endif
}

__device__ __forceinline__ void tdm_wait() {
#if __has_builtin(__builtin_amdgcn_s_wait_tensorcnt)
    __builtin_amdgcn_s_wait_tensorcnt(0);
#else
    asm volatile("s_wait_tensorcnt 0x0" ::: "memory");
#endif
}

// ---------------------------------------------------------------------------
// Kernel 0: zero BN partial sums (must run every launch: graph replays)
// ---------------------------------------------------------------------------
__global__ void zero_sums_kernel(float* __restrict__ sums) {
    int t = blockIdx.x * blockDim.x + threadIdx.x;
    if (t < 512) sums[t] = 0.0f;
}

// ---------------------------------------------------------------------------
// Kernel 1: KNN(k=3) interpolate. Wave-per-point (wave32).
// ---------------------------------------------------------------------------
__device__ __forceinline__ void top3_ins(float d, int i,
                                         float& d0, int& i0,
                                         float& d1, int& i1,
                                         float& d2, int& i2) {
    if (d < d2) {
        if (d < d1) {
            if (d < d0) { d2 = d1; i2 = i1; d1 = d0; i1 = i0; d0 = d; i0 = i; }
            else        { d2 = d1; i2 = i1; d1 = d;  i1 = i; }
        } else          { d2 = d;  i2 = i; }
    }
}

__global__ __launch_bounds__(256) void knn_interp_kernel(
    const float* __restrict__ points, const float* __restrict__ centroids,
    const float* __restrict__ cfeat, float* __restrict__ interp)
{
    __shared__ float sc[3 * M_];
    const int t  = threadIdx.x;
    const int gp = blockIdx.x * 8;
    const int b  = gp / N_;

    const float* cb = centroids + (size_t)b * 3 * M_;
    for (int i = t; i < 3 * M_; i += 256) sc[i] = cb[i];
    __syncthreads();

    const int w = t >> 5, l = t & 31;
    const int n = gp + w - b * N_;

    const float px = points[((size_t)b * 3 + 0) * N_ + n];
    const float py = points[((size_t)b * 3 + 1) * N_ + n];
    const float pz = points[((size_t)b * 3 + 2) * N_ + n];

    float d0 = 3.4e38f, d1 = 3.4e38f, d2 = 3.4e38f;
    int   i0 = 0, i1 = 0, i2 = 0;
    for (int m = l; m < M_; m += 32) {
        const float dx = px - sc[m];
        const float dy = py - sc[M_ + m];
        const float dz = pz - sc[2 * M_ + m];
        const float d = dx * dx + dy * dy + dz * dz;
        top3_ins(d, m, d0, i0, d1, i1, d2, i2);
    }
    for (int off = 16; off >= 1; off >>= 1) {
        const float e0 = __shfl_xor(d0, off, 32);
        const float e1 = __shfl_xor(d1, off, 32);
        const float e2 = __shfl_xor(d2, off, 32);
        const int   j0 = __shfl_xor(i0, off, 32);
        const int   j1 = __shfl_xor(i1, off, 32);
        const int   j2 = __shfl_xor(i2, off, 32);
        top3_ins(e0, j0, d0, i0, d1, i1, d2, i2);
        top3_ins(e1, j1, d0, i0, d1, i1, d2, i2);
        top3_ins(e2, j2, d0, i0, d1, i1, d2, i2);
    }
    float w0 = 1.0f / fmaxf(d0, 1e-16f);
    float w1 = 1.0f / fmaxf(d1, 1e-16f);
    float w2 = 1.0f / fmaxf(d2, 1e-16f);
    const float inv = 1.0f / (w0 + w1 + w2);
    w0 *= inv; w1 *= inv; w2 *= inv;

    const float* f0 = cfeat + ((size_t)b * M_ + i0) * CIN_;
    const float* f1 = cfeat + ((size_t)b * M_ + i1) * CIN_;
    const float* f2 = cfeat + ((size_t)b * M_ + i2) * CIN_;
    float* op = interp + ((size_t)b * N_ + n) * CIN_;

    #pragma unroll
    for (int jj = 0; jj < 2; ++jj) {
        const int c = l * 4 + jj * 128;
        const float4 a  = *(const float4*)(f0 + c);
        const float4 bb = *(const float4*)(f1 + c);
        const float4 cc = *(const float4*)(f2 + c);
        float4 r;
        r.x = w0 * a.x + w1 * bb.x + w2 * cc.x;
        r.y = w0 * a.y + w1 * bb.y + w2 * cc.y;
        r.z = w0 * a.z + w1 * bb.z + w2 * cc.z;
        r.w = w0 * a.w + w1 * bb.w + w2 * cc.w;
        *(float4*)(op + c) = r;
    }
}

__device__ __forceinline__ const float* gemm1_src(
    const float* pf, const float* interp, int b, int kc, int n0) {
    return (kc < CSK_)
        ? pf + ((size_t)b * CSK_ + kc) * N_ + n0
        : interp + (size_t)b * (size_t)N_ * CIN_ + (size_t)(kc - CSK_) * N_ + n0;
}

// ---------------------------------------------------------------------------
// Kernel 2: GEMM1 h1 = W1 @ [pf ; interp_view] + b1, plus BN partial sums.
//   256 threads = 8 waves; each wave owns TWO 16x64 output tiles (rows o0 and
//   o0+128) sharing B fragments -> 8 WMMA per 4 LDS fragment loads.
//   K staged by double-buffered TDM tiles (wave 0 issues, s_wait_tensorcnt).
// ---------------------------------------------------------------------------
__global__ __launch_bounds__(256) void gemm1_bn_kernel(
    const float* __restrict__ pf, const float* __restrict__ interp,
    const float* __restrict__ W1, const float* __restrict__ b1,
    float* __restrict__ h1, float* __restrict__ sums)
{
    __shared__ float xs[2][64 * 72];
    const int t  = threadIdx.x;
    const int b  = blockIdx.x >> 6;
    const int n0 = (blockIdx.x & 63) * 64;
    const int w  = t >> 5, l = t & 31;
    const int o0 = w * 16;
    const int lh = l & 15;
    const int hi = l >> 4;

    v8f acc[2][4];
    const v8f zero8 = {0.f, 0.f, 0.f, 0.f, 0.f, 0.f, 0.f, 0.f};
    #pragma unroll
    for (int s = 0; s < 2; ++s)
        #pragma unroll
        for (int f = 0; f < 4; ++f) acc[s][f] = zero8;

    const float* wrowA = W1 + (size_t)(o0 + lh) * C1_;
    const float* wrowB = W1 + (size_t)(o0 + 128 + lh) * C1_;
    const int NCH = C1_ / 64;   // 6 chunks

#if USE_TDM
    if (t < 32) tdm_load_tile_64x64(gemm1_src(pf, interp, b, 0, n0), xs[0]);
#endif
    for (int i = 0; i < NCH; ++i) {
        const int kc = i * 64;
#if USE_TDM
        if (t < 32) tdm_wait();
        asm volatile("" ::: "memory");
        __syncthreads();
        if (i + 1 < NCH && t < 32)
            tdm_load_tile_64x64(gemm1_src(pf, interp, b, kc + 64, n0), xs[(i + 1) & 1]);
        const float* xbuf = xs[i & 1];
#else
        {   // fallback: cooperative register staging (single buffer)
            const int ldr = t >> 2, ldq = t & 3;
            const float* src = gemm1_src(pf, interp, b, kc, n0) + (size_t)ldr * N_;
            float* dst = xs[0] + ldr * 72;
            #pragma unroll
            for (int s = 0; s < 4; ++s) {
                const int col = (ldq + 4 * s) * 4;
                *(float4*)(dst + col) = *(const float4*)(src + col);
            }
        }
        __syncthreads();
        const float* xbuf = xs[0];
#endif
        #pragma unroll
        for (int kk = 0; kk < 64; kk += 4) {
            const int ka = kc + kk + hi * 2;
            v2f a0; a0.x = wrowA[ka]; a0.y = wrowA[ka + 1];
            v2f a1; a1.x = wrowB[ka]; a1.y = wrowB[ka + 1];
            const int kb = kk + hi * 2;
            #pragma unroll
            for (int f = 0; f < 4; ++f) {
                const int colb = f * 16 + lh;
                v2f bf;
                bf.x = xbuf[kb * 72 + colb];
                bf.y = xbuf[(kb + 1) * 72 + colb];
                acc[0][f] = __builtin_amdgcn_wmma_f32_16x16x4_f32(
                    false, a0, false, bf, (short)0, acc[0][f], false, false);
                acc[1][f] = __builtin_amdgcn_wmma_f32_16x16x4_f32(
                    false, a1, false, bf, (short)0, acc[1][f], false, false);
            }
        }
#if !USE_TDM
        __syncthreads();
#endif
    }

    #pragma unroll
    for (int set = 0; set < 2; ++set) {
        #pragma unroll
        for (int j = 0; j < 8; ++j) {
            const int ch = set * 128 + o0 + j + hi * 8;
            const float bias = b1[ch];
            float s = 0.f, q = 0.f;
            #pragma unroll
            for (int f = 0; f < 4; ++f) {
                const float v = acc[set][f][j] + bias;
                h1[((size_t)b * COUT_ + ch) * N_ + n0 + f * 16 + lh] = v;
                s += v; q += v * v;
            }
            for (int off = 8; off >= 1; off >>= 1) {
                s += __shfl_xor(s, off, 32);
                q += __shfl_xor(q, off, 32);
            }
            if (lh == 0) {
                atomicAdd(&sums[ch], s);
                atomicAdd(&sums[256 + ch], q);
            }
        }
    }
}

// ---------------------------------------------------------------------------
// Kernel 3: fold BN stats + gamma/beta into per-channel scale & bias.
// ---------------------------------------------------------------------------
__global__ void bn_finalize_kernel(const float* __restrict__ sums,
                                   const float* __restrict__ gamma,
                                   const float* __restrict__ beta,
                                   float* __restrict__ scalebias)
{
    const int c = threadIdx.x;            // 256 threads
    const float invN = 1.0f / (float)(B_ * N_);
    const float mean = sums[c] * invN;
    const float var  = sums[256 + c] * invN - mean * mean;
    const float sc   = gamma[c] * rsqrtf(var + BN_EPS_);
    scalebias[c]       = sc;
    scalebias[256 + c] = beta[c] - mean * sc;
}

// ---------------------------------------------------------------------------
// Kernel 4: out = W2 @ relu(BN(h1)) + b2.
//   Double-buffered TDM loads raw h1 tiles; BN+ReLU applied by an in-LDS
//   transform pass between two barriers; two o-tiles per wave.
// ---------------------------------------------------------------------------
__global__ __launch_bounds__(256) void gemm2_kernel(
    const float* __restrict__ h1, const float* __restrict__ W2,
    const float* __restrict__ b2, const float* __restrict__ scalebias,
    float* __restrict__ out)
{
    __shared__ float xs[2][64 * 72];
    const int t  = threadIdx.x;
    const int b  = blockIdx.x >> 6;
    const int n0 = (blockIdx.x & 63) * 64;
    const int w  = t >> 5, l = t & 31;
    const int o0 = w * 16;
    const int lh = l & 15;
    const int hi = l >> 4;

    v8f acc[2][4];
    const v8f zero8 = {0.f, 0.f, 0.f, 0.f, 0.f, 0.f, 0.f, 0.f};
    #pragma unroll
    for (int s = 0; s < 2; ++s)
        #pragma unroll
        for (int f = 0; f < 4; ++f) acc[s][f] = zero8;

    const float* wrowA = W2 + (size_t)(o0 + lh) * COUT_;
    const float* wrowB = W2 + (size_t)(o0 + 128 + lh) * COUT_;
    const int ldr = t >> 2;
    const int ldq = t & 3;
    const int NCH = COUT_ / 64;   // 4 chunks
    const float* h1b = h1 + ((size_t)b * COUT_) * N_ + n0;

#if USE_TDM
    if (t < 32) tdm_load_tile_64x64(h1b, xs[0]);
#endif
    for (int i = 0; i < NCH; ++i) {
        const int kc = i * 64;
        const int c  = kc + ldr;
        const float scv = scalebias[c];
        const float biv = scalebias[256 + c];
#if USE_TDM
        if (t < 32) tdm_wait();
        asm volatile("" ::: "memory");
        __syncthreads();
        if (i + 1 < NCH && t < 32)
            tdm_load_tile_64x64(h1b + (size_t)(kc + 64) * N_, xs[(i + 1) & 1]);
        float* xbuf = xs[i & 1];
        {   // in-LDS BN + ReLU transform (each thread owns its 16 floats)
            float* dst = xbuf + ldr * 72;
            #pragma unroll
            for (int s = 0; s < 4; ++s) {
                const int col = (ldq + 4 * s) * 4;
                float4 v = *(float4*)(dst + col);
                v.x = fmaxf(v.x * scv + biv, 0.f);
                v.y = fmaxf(v.y * scv + biv, 0.f);
                v.z = fmaxf(v.z * scv + biv, 0.f);
                v.w = fmaxf(v.w * scv + biv, 0.f);
                *(float4*)(dst + col) = v;
            }
        }
        __syncthreads();
#else
        {   // fallback: stage relu(h1*scale+bias) through registers
            const float* src = h1b + (size_t)c * N_;
            float* dst = xs[0] + ldr * 72;
            #pragma unroll
            for (int s = 0; s < 4; ++s) {
                const int col = (ldq + 4 * s) * 4;
                float4 v = *(const float4*)(src + col);
                v.x = fmaxf(v.x * scv + biv, 0.f);
                v.y = fmaxf(v.y * scv + biv, 0.f);
                v.z = fmaxf(v.z * scv + biv, 0.f);
                v.w = fmaxf(v.w * scv + biv, 0.f);
                *(float4*)(dst + col) = v;
            }
        }
        __syncthreads();
        float* xbuf = xs[0];
#endif
        #pragma unroll
        for (int kk = 0; kk < 64; kk += 4) {
            const int ka = kc + kk + hi * 2;
            v2f a0; a0.x = wrowA[ka]; a0.y = wrowA[ka + 1];
            v2f a1; a1.x = wrowB[ka]; a1.y = wrowB[ka + 1];
            const int kb = kk + hi * 2;
            #pragma unroll
            for (int f = 0; f < 4; ++f) {
                const int colb = f * 16 + lh;
                v2f bf;
                bf.x = xbuf[kb * 72 + colb];
                bf.y = xbuf[(kb + 1) * 72 + colb];
                acc[0][f] = __builtin_amdgcn_wmma_f32_16x16x4_f32(
                    false, a0, false, bf, (short)0, acc[0][f], false, false);
                acc[1][f] = __builtin_amdgcn_wmma_f32_16x16x4_f32(
                    false, a1, false, bf, (short)0, acc[1][f], false, false);
            }
        }
#if !USE_TDM
        __syncthreads();
#endif
    }

    #pragma unroll
    for (int set = 0; set < 2; ++set) {
        #pragma unroll
        for (int j = 0; j < 8; ++j) {
            const int ch = set * 128 + o0 + j + hi * 8;
            const float bias = b2[ch];
            #pragma unroll
            for (int f = 0; f < 4; ++f) {
                __builtin_nontemporal_store(
                    acc[set][f][j] + bias,
                    &out[((size_t)b * COUT_ + ch) * N_ + n0 + f * 16 + lh]);
            }
        }
    }
}

// ---------------------------------------------------------------------------
extern "C" void kernel_launch(void* const* d_in, const int* in_sizes, int n_in,
                              void* d_out, int out_size, void* d_ws, size_t ws_size,
                              hipStream_t stream) {
    const float* points   = (const float*)d_in[0];
    const float* pfeat    = (const float*)d_in[1];
    const float* cents    = (const float*)d_in[2];
    const float* cfeat    = (const float*)d_in[3];
    const float* W1       = (const float*)d_in[4];
    const float* b1       = (const float*)d_in[5];
    const float* gamma    = (const float*)d_in[6];
    const float* beta     = (const float*)d_in[7];
    const float* W2       = (const float*)d_in[8];
    const float* b2       = (const float*)d_in[9];
    float* out = (float*)d_out;

    float* ws        = (float*)d_ws;
    float* interp    = ws;                                   // B*N*CIN
    float* h1        = interp + (size_t)B_ * N_ * CIN_;      // B*COUT*N
    float* sums      = h1 + (size_t)B_ * COUT_ * N_;         // 512
    float* scalebias = sums + 512;                           // 512

    zero_sums_kernel<<<2, 256, 0, stream>>>(sums);

    knn_interp_kernel<<<(B_ * N_) / 8, 256, 0, stream>>>(points, cents, cfeat, interp);

    gemm1_bn_kernel<<<B_ * (N_ / 64), 256, 0, stream>>>(pfeat, interp, W1, b1, h1, sums);

    bn_finalize_kernel<<<1, 256, 0, stream>>>(sums, gamma, beta, scalebias);

    gemm2_kernel<<<B_ * (N_ / 64), 256, 0, stream>>>(h1, W2, b2, scalebias, out);
}